// FuncSelfAttention_18262200942798
// MI455X (gfx1250) — compile-verified
//
#include <hip/hip_runtime.h>

// ---------------------------------------------------------------------------
// FuncSelfAttention (spectral FNO attention) for MI455X / gfx1250.
// All heavy GEMMs use V_WMMA_F32_16X16X4_F32 (wave32, 16x16 tiles, K=4 steps).
// ---------------------------------------------------------------------------

typedef __attribute__((ext_vector_type(2))) float v2f;
typedef __attribute__((ext_vector_type(8))) float v8f;

// 8-arg pattern: (neg_a, A, neg_b, B, c_mod, C, reuse_a, reuse_b)
#define WMMA_F32(a, b, c) \
  __builtin_amdgcn_wmma_f32_16x16x4_f32(false, (a), false, (b), (short)0, (c), false, false)

static constexpr int EMBED = 128, HEADS = 4, HEAD_DIM = 32, MODES = 16;
static constexpr int B_ = 4, S_ = 64, H_ = 32, W_ = 32;
static constexpr int PIX = H_ * W_;          // 1024
static constexpr int BS = B_ * S_;           // 256
static constexpr int NM = MODES * MODES;     // 256 modes
static constexpr int C3 = 3 * EMBED;         // 384
static constexpr int KSPLIT = 8;             // K-split for attention scores
static constexpr float TWO_PI = 6.283185307179586f;

// --------------------------------------------------------------------------
// Weight re-layout: w[Cin][Cout][16][16] -> wt[mode][k/2][Cout][2]
// (pair-interleave adjacent K so a lane's B fragment is one aligned b64 load)
// --------------------------------------------------------------------------
__global__ void k_transpose_w(const float* __restrict__ w, float* __restrict__ wt,
                              int Cin, int Cout) {
  int idx = blockIdx.x * blockDim.x + threadIdx.x;
  int total = Cin * Cout * NM;
  if (idx >= total) return;
  int p = idx & 1;
  int r = idx >> 1;
  int o = r % Cout;
  int r2 = r / Cout;
  int kk = r2 % (Cin >> 1);
  int mode = r2 / (Cin >> 1);
  int ci = 2 * kk + p;
  wt[idx] = w[((size_t)ci * Cout + o) * NM + mode];
}

// --------------------------------------------------------------------------
// Swin-v2 log-CPB bias: bias[h][s][t], s,t over 8x8 subdomain grid.
// --------------------------------------------------------------------------
__global__ void k_cpb_bias(const float* __restrict__ w1, const float* __restrict__ b1,
                           const float* __restrict__ w2, const float* __restrict__ b2,
                           float* __restrict__ bias) {
  int id = blockIdx.x * blockDim.x + threadIdx.x;  // 4096 = S*S
  if (id >= S_ * S_) return;
  int s = id >> 6, t = id & 63;
  float r0 = (float)((s >> 3) - (t >> 3));
  float r1 = (float)((s & 7) - (t & 7));
  r0 = (r0 >= 0.f ? 1.f : -1.f) * log1pf(fabsf(r0));
  r1 = (r1 >= 0.f ? 1.f : -1.f) * log1pf(fabsf(r1));
  float a0 = 0.f, a1 = 0.f, a2 = 0.f, a3 = 0.f;
  for (int j = 0; j < EMBED; ++j) {
    float hj = fmaxf(r0 * w1[j] + r1 * w1[EMBED + j] + b1[j], 0.f);
    a0 += hj * w2[j * HEADS + 0];
    a1 += hj * w2[j * HEADS + 1];
    a2 += hj * w2[j * HEADS + 2];
    a3 += hj * w2[j * HEADS + 3];
  }
  bias[0 * 4096 + id] = a0 + b2[0];
  bias[1 * 4096 + id] = a1 + b2[1];
  bias[2 * 4096 + id] = a2 + b2[2];
  bias[3 * 4096 + id] = a3 + b2[3];
}

// --------------------------------------------------------------------------
// Truncated forward rDFT2: one 32x32 image -> 16x16 complex modes.
// Output layout is mode-major: xf[(mode*BS + bs)*Cin + ci]  (feeds WMMA GEMM)
// --------------------------------------------------------------------------
__global__ void __launch_bounds__(256) k_dft_fwd(const float* __restrict__ img,
                                                 float* __restrict__ xfr,
                                                 float* __restrict__ xfi, int Cin) {
  __shared__ float X[PIX];
  __shared__ float Ar[32 * MODES], Ai[32 * MODES];
  __shared__ float ct[32], st[32];
  int tid = threadIdx.x;
  int im = blockIdx.x;
  int bs = im / Cin, ci = im % Cin;
  if (tid < 32) {
    float sv, cv;
    __sincosf(TWO_PI * (float)tid / 32.f, &sv, &cv);
    ct[tid] = cv; st[tid] = sv;
  }
  const float* p = img + (size_t)im * PIX;
  for (int i = tid; i < PIX; i += 256) X[i] = p[i];
  __syncthreads();
  // along y: A[x][ky] = sum_y X[x][y] * e^{-i 2pi ky y / 32}
  for (int idx = tid; idx < 32 * MODES; idx += 256) {
    int x = idx >> 4, ky = idx & 15;
    float sr = 0.f, si = 0.f;
    const float* row = &X[x * 32];
    for (int y = 0; y < 32; ++y) {
      int ph = (ky * y) & 31;
      float v = row[y];
      sr += v * ct[ph];
      si -= v * st[ph];
    }
    Ar[idx] = sr; Ai[idx] = si;
  }
  __syncthreads();
  // along x: Z[kx][ky] = sum_x e^{-i 2pi kx x/32} A[x][ky]
  int kx = tid >> 4, ky = tid & 15;
  float zr = 0.f, zi = 0.f;
  for (int x = 0; x < 32; ++x) {
    int ph = (kx * x) & 31;
    float c = ct[ph], s = st[ph];
    float ar = Ar[x * 16 + ky], ai = Ai[x * 16 + ky];
    zr += c * ar + s * ai;
    zi += c * ai - s * ar;
  }
  size_t dst = ((size_t)tid * BS + bs) * Cin + ci;  // mode == tid
  xfr[dst] = zr;
  xfi[dst] = zi;
}

// --------------------------------------------------------------------------
// Per-mode complex GEMM:  OF[bs][o] = sum_ci XF[bs][ci] * W[ci][o]  (complex)
// Re = Xr*Wr - Xi*Wi ; Im = Xr*Wi + Xi*Wr  -> 4 f32 WMMA per K=4 chunk.
// grid = (Cout/128, BS/16, NM), block = 256 (8 waves; 16 N-cols per wave).
// B operands come as aligned b64 pairs from the interleaved weight layout.
// Output layout: of[(bs*Cout + o)*NM + mode]  (feeds inverse DFT)
// --------------------------------------------------------------------------
__global__ void __launch_bounds__(256) k_spectral_mm(
    const float* __restrict__ xfr, const float* __restrict__ xfi,
    const float* __restrict__ wtr, const float* __restrict__ wti,
    float* __restrict__ ofr, float* __restrict__ ofi, int Cin, int Cout) {
  __shared__ float Axr[16 * 128];
  __shared__ float Axi[16 * 128];
  int tid = threadIdx.x;
  int mode = blockIdx.z;
  int mbase = blockIdx.y * 16;
  for (int i = tid; i < 16 * Cin; i += 256) {
    int r = i / Cin, k = i % Cin;
    size_t src = ((size_t)mode * BS + (mbase + r)) * Cin + k;
    Axr[i] = xfr[src];
    Axi[i] = xfi[src];
  }
  __syncthreads();
  int wave = tid >> 5, lane = tid & 31;
  int nb = blockIdx.x * 128 + wave * 16;
  int m = lane & 15, n = lane & 15;
  int khalf = (lane >> 4) * 2;  // lanes 16-31 hold K+2,K+3 halves
  v8f accR = {0.f, 0.f, 0.f, 0.f, 0.f, 0.f, 0.f, 0.f};
  v8f accI = accR;
  // interleaved layout: wt[mode][kk][Cout][2] viewed as v2f[mode][kk][Cout]
  const v2f* Br = (const v2f*)wtr + (size_t)mode * (Cin >> 1) * Cout + nb + n;
  const v2f* Bi = (const v2f*)wti + (size_t)mode * (Cin >> 1) * Cout + nb + n;
  for (int k0 = 0; k0 < Cin; k0 += 4) {
    int kk = (k0 + khalf) >> 1;
    if (k0 + 16 < Cin) {
      __builtin_prefetch((const float*)(Br + (size_t)(kk + 8) * Cout), 0, 1);
      __builtin_prefetch((const float*)(Bi + (size_t)(kk + 8) * Cout), 0, 1);
    }
    int ka = k0 + khalf;
    v2f ar = {Axr[m * 128 + ka], Axr[m * 128 + ka + 1]};
    v2f ai = {Axi[m * 128 + ka], Axi[m * 128 + ka + 1]};
    v2f ain = -ai;  // f32 WMMA has no A-negate; negate imaginary fragment in VALU
    v2f br = Br[(size_t)kk * Cout];
    v2f bi = Bi[(size_t)kk * Cout];
    accR = WMMA_F32(ar, br, accR);
    accR = WMMA_F32(ain, bi, accR);
    accI = WMMA_F32(ar, bi, accI);
    accI = WMMA_F32(ai, br, accI);
  }
  int rowoff = (lane < 16) ? 0 : 8;  // C layout: VGPR r -> rows r / r+8
  int col = nb + (lane & 15);
#pragma unroll
  for (int r = 0; r < 8; ++r) {
    int row = mbase + r + rowoff;
    size_t dst = ((size_t)row * Cout + col) * NM + mode;
    ofr[dst] = accR[r];
    ofi[dst] = accI[r];
  }
}

// --------------------------------------------------------------------------
// Inverse: 16x16 complex modes -> real 32x32 (irfft2 semantics, 1/(H*W)),
// optional residual add (qkv path tiles seq 3x along channels).
// --------------------------------------------------------------------------
__global__ void __launch_bounds__(256) k_idft(const float* __restrict__ ofr,
                                              const float* __restrict__ ofi,
                                              const float* __restrict__ res,
                                              float* __restrict__ dst, int Cout) {
  __shared__ float Zr[NM], Zi[NM];
  __shared__ float Tr[32 * MODES], Ti[32 * MODES];
  __shared__ float ct[32], st[32];
  int tid = threadIdx.x;
  int blk = blockIdx.x;
  int bs = blk / Cout, o = blk % Cout;
  if (tid < 32) {
    float sv, cv;
    __sincosf(TWO_PI * (float)tid / 32.f, &sv, &cv);
    ct[tid] = cv; st[tid] = sv;
  }
  size_t src = ((size_t)bs * Cout + o) * NM + tid;
  Zr[tid] = ofr[src];
  Zi[tid] = ofi[src];
  __syncthreads();
  // inverse along x (full complex ifft, only 16 coeffs nonzero)
  for (int idx = tid; idx < 32 * MODES; idx += 256) {
    int x = idx >> 4, ky = idx & 15;
    float sr = 0.f, si = 0.f;
    for (int kx = 0; kx < 16; ++kx) {
      int ph = (kx * x) & 31;
      float c = ct[ph], s = st[ph];
      float zr = Zr[kx * 16 + ky], zi = Zi[kx * 16 + ky];
      sr += c * zr - s * zi;
      si += c * zi + s * zr;
    }
    Tr[idx] = sr; Ti[idx] = si;
  }
  __syncthreads();
  // irfft along y (Hermitian half-spectrum, ky=0..15 only, ky=16 zero)
  const float inv = 1.0f / (float)PIX;
  for (int idx = tid; idx < PIX; idx += 256) {
    int x = idx >> 5, y = idx & 31;
    float acc = Tr[x * 16];  // ky=0 term (imag dropped, per irfft)
    for (int ky = 1; ky < 16; ++ky) {
      int ph = (ky * y) & 31;
      acc += 2.f * (Tr[x * 16 + ky] * ct[ph] - Ti[x * 16 + ky] * st[ph]);
    }
    acc *= inv;
    if (res) acc += res[((size_t)bs * EMBED + (o % EMBED)) * PIX + idx];
    dst[((size_t)bs * Cout + o) * PIX + idx] = acc;
  }
}

// --------------------------------------------------------------------------
// Partial Q.K^T: split the 32768-deep K dim KSPLIT ways for occupancy.
// grid = (KSPLIT, 16 bh); block = 256; wave owns 2 of the 16 output tiles.
// part[(bh*KSPLIT + ks)][64][64] holds raw partial dot products.
// --------------------------------------------------------------------------
__global__ void __launch_bounds__(256) k_attn_scores_part(const float* __restrict__ qkv,
                                                          float* __restrict__ part) {
  __shared__ float Qt[64 * 64];
  __shared__ float Kt[64 * 64];
  int tid = threadIdx.x;
  int ks = blockIdx.x;
  int bh = blockIdx.y;
  int b = bh >> 2, h = bh & 3;
  const size_t rowstride = (size_t)C3 * PIX;  // 393216 floats between s rows
  const float* qp = qkv + ((size_t)b * S_ * C3 + h * HEAD_DIM) * PIX;
  const float* kp = qkv + ((size_t)b * S_ * C3 + EMBED + h * HEAD_DIM) * PIX;
  int wave = tid >> 5, lane = tid & 31;
  int mt = wave & 3;
  int n0 = ((wave >> 2) * 2) * 16, n1 = n0 + 16;
  int m = lane & 15, n = lane & 15;
  int khalf = (lane >> 4) * 2;
  v8f acc0 = {0.f, 0.f, 0.f, 0.f, 0.f, 0.f, 0.f, 0.f};
  v8f acc1 = acc0;
  const int kspan = HEAD_DIM * PIX / KSPLIT;  // 4096
  const int kbeg = ks * kspan;
  for (int kb = kbeg; kb < kbeg + kspan; kb += 64) {
    __syncthreads();
    for (int i = tid; i < 4096; i += 256) {
      int r = i >> 6, c = i & 63;
      Qt[i] = qp[(size_t)r * rowstride + kb + c];
      Kt[i] = kp[(size_t)r * rowstride + kb + c];
    }
    __syncthreads();
#pragma unroll
    for (int k0 = 0; k0 < 64; k0 += 4) {
      int ka = k0 + khalf;
      v2f ar = {Qt[(mt * 16 + m) * 64 + ka], Qt[(mt * 16 + m) * 64 + ka + 1]};
      v2f b0 = {Kt[(n0 + n) * 64 + ka], Kt[(n0 + n) * 64 + ka + 1]};
      v2f b1 = {Kt[(n1 + n) * 64 + ka], Kt[(n1 + n) * 64 + ka + 1]};
      acc0 = WMMA_F32(ar, b0, acc0);
      acc1 = WMMA_F32(ar, b1, acc1);
    }
  }
  int rowoff = (lane < 16) ? 0 : 8;
  float* pp = part + ((size_t)bh * KSPLIT + ks) * 4096;
#pragma unroll
  for (int r = 0; r < 8; ++r) {
    int row = mt * 16 + r + rowoff;
    pp[row * 64 + n0 + (lane & 15)] = acc0[r];
    pp[row * 64 + n1 + (lane & 15)] = acc1[r];
  }
}

// --------------------------------------------------------------------------
// Reduce K-split partials, scale, add CPB bias, row softmax -> att[bh][64][64].
// --------------------------------------------------------------------------
__global__ void __launch_bounds__(256) k_attn_softmax(const float* __restrict__ part,
                                                      const float* __restrict__ bias,
                                                      float* __restrict__ att) {
  __shared__ float St[64 * 64];
  __shared__ float rmax[64], rsum[64];
  int tid = threadIdx.x;
  int bh = blockIdx.x;
  int h = bh & 3;
  const float scale = 0.00552427172802f;  // 1/sqrt(32*32*32)
  for (int i = tid; i < 4096; i += 256) {
    float acc = 0.f;
#pragma unroll
    for (int ks = 0; ks < KSPLIT; ++ks)
      acc += part[((size_t)bh * KSPLIT + ks) * 4096 + i];
    int row = i >> 6, col = i & 63;
    St[i] = acc * scale + bias[(h * 64 + row) * 64 + col];
  }
  __syncthreads();
  if (tid < 64) {
    float mx = -3.4e38f;
    for (int t = 0; t < 64; ++t) mx = fmaxf(mx, St[tid * 64 + t]);
    float sm = 0.f;
    for (int t = 0; t < 64; ++t) sm += __expf(St[tid * 64 + t] - mx);
    rmax[tid] = mx;
    rsum[tid] = sm;
  }
  __syncthreads();
  for (int i = tid; i < 4096; i += 256) {
    int row = i >> 6;
    att[(size_t)bh * 4096 + i] = __expf(St[i] - rmax[row]) / rsum[row];
  }
}

// --------------------------------------------------------------------------
// sa[b,h,s,d] = sum_t att[b,h,s,t] * v[b,h,t,d]  (M=64, K=64, N=32768).
// grid = (256 N-chunks, 16 bh); wave holds 4 accumulators (all 4 M-tiles).
// Output re-flattened to sa[bs][128 ch][1024 pix].
// --------------------------------------------------------------------------
__global__ void __launch_bounds__(256) k_attn_apply(const float* __restrict__ att,
                                                    const float* __restrict__ qkv,
                                                    float* __restrict__ sa) {
  __shared__ float At[4096];
  int tid = threadIdx.x;
  int bh = blockIdx.y;
  int b = bh >> 2, h = bh & 3;
  for (int i = tid; i < 4096; i += 256) At[i] = att[(size_t)bh * 4096 + i];
  __syncthreads();
  int wave = tid >> 5, lane = tid & 31;
  int nb = blockIdx.x * 128 + wave * 16;
  int m = lane & 15, n = lane & 15;
  int khalf = (lane >> 4) * 2;
  const size_t rowstride = (size_t)C3 * PIX;
  const float* vp = qkv + ((size_t)b * S_ * C3 + 2 * EMBED + h * HEAD_DIM) * PIX;
  v8f z = {0.f, 0.f, 0.f, 0.f, 0.f, 0.f, 0.f, 0.f};
  v8f acc[4];
#pragma unroll
  for (int mt = 0; mt < 4; ++mt) acc[mt] = z;
#pragma unroll 4
  for (int k0 = 0; k0 < 64; k0 += 4) {
    int ka = k0 + khalf;
    v2f bf = {vp[(size_t)ka * rowstride + nb + n],
              vp[(size_t)(ka + 1) * rowstride + nb + n]};
#pragma unroll
    for (int mt = 0; mt < 4; ++mt) {
      v2f ar = {At[(mt * 16 + m) * 64 + ka], At[(mt * 16 + m) * 64 + ka + 1]};
      acc[mt] = WMMA_F32(ar, bf, acc[mt]);
    }
  }
  int rowoff = (lane < 16) ? 0 : 8;
  int col = nb + (lane & 15);
  int hd = col >> 10, pix = col & 1023;
#pragma unroll
  for (int mt = 0; mt < 4; ++mt) {
#pragma unroll
    for (int r = 0; r < 8; ++r) {
      int s = mt * 16 + r + rowoff;
      sa[((size_t)(b * S_ + s) * EMBED + h * HEAD_DIM + hd) * PIX + pix] = acc[mt][r];
    }
  }
}

// --------------------------------------------------------------------------
extern "C" void kernel_launch(void* const* d_in, const int* in_sizes, int n_in,
                              void* d_out, int out_size, void* d_ws, size_t ws_size,
                              hipStream_t stream) {
  (void)in_sizes; (void)n_in; (void)out_size; (void)ws_size;
  const float* seq    = (const float*)d_in[0];
  const float* qkv_wr = (const float*)d_in[1];
  const float* qkv_wi = (const float*)d_in[2];
  const float* out_wr = (const float*)d_in[3];
  const float* out_wi = (const float*)d_in[4];
  const float* cpb_w1 = (const float*)d_in[5];
  const float* cpb_b1 = (const float*)d_in[6];
  const float* cpb_w2 = (const float*)d_in[7];
  const float* cpb_b2 = (const float*)d_in[8];
  float* out = (float*)d_out;

  float* W = (float*)d_ws;
  size_t off = 0;
  float* BIAS = W + off; off += (size_t)HEADS * S_ * S_;             // 16K
  float* ATT  = W + off; off += (size_t)B_ * HEADS * S_ * S_;        // 64K
  float* PART = W + off; off += (size_t)B_ * HEADS * KSPLIT * S_ * S_; // 512K
  float* WTQR = W + off; off += (size_t)NM * EMBED * C3;             // 12.6M
  float* WTQI = W + off; off += (size_t)NM * EMBED * C3;
  float* WTOR = W + off; off += (size_t)NM * EMBED * EMBED;          // 4.2M
  float* WTOI = W + off; off += (size_t)NM * EMBED * EMBED;
  float* XFR  = W + off; off += (size_t)NM * BS * EMBED;             // 8.4M (both DFT passes)
  float* XFI  = W + off; off += (size_t)NM * BS * EMBED;
  float* OFR  = W + off; off += (size_t)BS * C3 * NM;                // 25.2M (out-stage reuses prefix)
  float* OFI  = W + off; off += (size_t)BS * C3 * NM;
  float* QKV  = W + off; off += (size_t)BS * C3 * PIX;               // 100.7M
  float* SA   = W + off; off += (size_t)BS * EMBED * PIX;            // 33.6M

  // 1. weight re-layout to mode-major, K-pair interleaved
  {
    int totq = NM * EMBED * C3;
    k_transpose_w<<<(totq + 255) / 256, 256, 0, stream>>>(qkv_wr, WTQR, EMBED, C3);
    k_transpose_w<<<(totq + 255) / 256, 256, 0, stream>>>(qkv_wi, WTQI, EMBED, C3);
    int toto = NM * EMBED * EMBED;
    k_transpose_w<<<(toto + 255) / 256, 256, 0, stream>>>(out_wr, WTOR, EMBED, EMBED);
    k_transpose_w<<<(toto + 255) / 256, 256, 0, stream>>>(out_wi, WTOI, EMBED, EMBED);
  }
  // 2. relative position bias MLP
  k_cpb_bias<<<16, 256, 0, stream>>>(cpb_w1, cpb_b1, cpb_w2, cpb_b2, BIAS);
  // 3. qkv spectral conv
  k_dft_fwd<<<BS * EMBED, 256, 0, stream>>>(seq, XFR, XFI, EMBED);
  k_spectral_mm<<<dim3(C3 / 128, BS / 16, NM), 256, 0, stream>>>(
      XFR, XFI, WTQR, WTQI, OFR, OFI, EMBED, C3);
  k_idft<<<BS * C3, 256, 0, stream>>>(OFR, OFI, seq, QKV, C3);
  // 4. attention
  k_attn_scores_part<<<dim3(KSPLIT, B_ * HEADS), 256, 0, stream>>>(QKV, PART);
  k_attn_softmax<<<B_ * HEADS, 256, 0, stream>>>(PART, BIAS, ATT);
  k_attn_apply<<<dim3(HEAD_DIM * PIX / 128, B_ * HEADS), 256, 0, stream>>>(ATT, QKV, SA);
  // 5. output spectral conv (no residual)
  k_dft_fwd<<<BS * EMBED, 256, 0, stream>>>(SA, XFR, XFI, EMBED);
  k_spectral_mm<<<dim3(1, BS / 16, NM), 256, 0, stream>>>(
      XFR, XFI, WTOR, WTOI, OFR, OFI, EMBED, EMBED);
  k_idft<<<BS * EMBED, 256, 0, stream>>>(OFR, OFI, nullptr, out, EMBED);
}